// ProposalGenerator_43757126812082
// MI455X (gfx1250) — compile-verified
//
#include <hip/hip_runtime.h>
#include <hip/hip_bf16.h>
#include <math.h>

#define HID  256
#define PROJ 256
#define DYN  64
#define HW   6400
#define BSZ  16
#define NQ   100
#define HH   80
#define WW   80
#define NPAD 112          // n padded to 7 WMMA tiles of 16
#define QPAD 260          // LDS row stride (floats): 260%64=4 -> conflict-free b64 A loads
#define KTILES 4          // k-subtiles (16 rows each) per block -> 64 rows

typedef float v2f __attribute__((ext_vector_type(2)));
typedef float v8f __attribute__((ext_vector_type(8)));

// ---------------------------------------------------------------------------
// K1: fs_feat[b,n,:] = (tanh(relu(fs_proj@W1+b1)@W2+b2)+1)*fs_proj,
//     fs_proj = sf@Ws+bsp ; also d[b,n] = bq . fs_feat[b,n,:]
// grid (NQ, BS), block 256
// ---------------------------------------------------------------------------
__global__ void __launch_bounds__(256)
fs_feat_kernel(const float* __restrict__ support,   // [NQ,BS,HID]
               const float* __restrict__ Ws, const float* __restrict__ bsp,
               const float* __restrict__ W1, const float* __restrict__ b1,
               const float* __restrict__ W2, const float* __restrict__ b2,
               const float* __restrict__ bq,
               float* __restrict__ fsfeat,          // [BS,NQ,PROJ]
               float* __restrict__ dws)             // [BS,NQ]
{
  __shared__ float sfrow[HID];
  __shared__ float fsrow[PROJ];
  __shared__ float hid[DYN];
  __shared__ float red[256];
  const int n = blockIdx.x, b = blockIdx.y, p = threadIdx.x;

  sfrow[p] = support[(n * BSZ + b) * HID + p];
  __syncthreads();

  float fsp = bsp[p];
#pragma unroll 4
  for (int c = 0; c < HID; ++c) fsp = fmaf(sfrow[c], Ws[c * PROJ + p], fsp);
  fsrow[p] = fsp;
  __syncthreads();

  if (p < DYN) {
    float hh = b1[p];
#pragma unroll 4
    for (int c = 0; c < PROJ; ++c) hh = fmaf(fsrow[c], W1[c * DYN + p], hh);
    hid[p] = fmaxf(hh, 0.0f);
  }
  __syncthreads();

  float pat = b2[p];
#pragma unroll 4
  for (int j = 0; j < DYN; ++j) pat = fmaf(hid[j], W2[j * HID + p], pat);
  pat = tanhf(pat);
  const float ff = (pat + 1.0f) * fsp;
  fsfeat[((size_t)b * NQ + n) * PROJ + p] = ff;

  red[p] = bq[p] * ff;
  __syncthreads();
  for (int s = 128; s > 0; s >>= 1) {
    if (p < s) red[p] += red[p + s];
    __syncthreads();
  }
  if (p == 0) dws[b * NQ + n] = red[0];
}

// ---------------------------------------------------------------------------
// K2: Mt2[b][c/4][n][c%4] = sum_p Wq[c,p] * fs_feat[b,n,p]
//     (K-interleaved layout: the 2 K-values a WMMA lane needs are adjacent ->
//      single global_load_b64 B-fragment in K3; n padded with zeros to 112)
// grid (HID, BS), block 128
// ---------------------------------------------------------------------------
__global__ void __launch_bounds__(128)
mt_kernel(const float* __restrict__ Wq,       // [HID,PROJ]
          const float* __restrict__ fsfeat,   // [BS,NQ,PROJ]
          float* __restrict__ Mt)             // [BS,HID/4,NPAD,4]
{
  __shared__ float wqrow[PROJ];
  const int c = blockIdx.x, b = blockIdx.y, t = threadIdx.x;
  for (int i = t; i < PROJ; i += 128) wqrow[i] = Wq[c * PROJ + i];
  __syncthreads();
  if (t < NPAD) {
    float acc = 0.0f;
    if (t < NQ) {
      const float* f = &fsfeat[((size_t)b * NQ + t) * PROJ];
#pragma unroll 4
      for (int p = 0; p < PROJ; ++p) acc = fmaf(wqrow[p], f[p], acc);
    }
    Mt[(((size_t)b * (HID / 4) + (c >> 2)) * NPAD + t) * 4 + (c & 3)] = acc;
  }
}

// ---------------------------------------------------------------------------
// K3: sim[b,n,k] = sum_c Q[k,b,c] * Mt[b,c,n] + d[b,n]   via V_WMMA_F32_16X16X4_F32
// grid (HW/64, BS), block 224 (7 waves; wave w owns n-tile w across 4 k-subtiles)
// Per c-step: 1 global_load_b64 (B, reused 4x) + 4 ds_load_b64 (A) + 4 wmma.
// ---------------------------------------------------------------------------
__global__ void __launch_bounds__(224)
sim_wmma_kernel(const float* __restrict__ query,  // [HW,BS,HID]
                const float* __restrict__ Mt,     // [BS,HID/4,NPAD,4]
                const float* __restrict__ dws,    // [BS,NQ]
                float* __restrict__ sim)          // [BS,NQ,HW]
{
  __shared__ float ldsQ[16 * KTILES * QPAD];      // 64 rows x 260 floats = 65KB
  const int k0 = blockIdx.x * (16 * KTILES);
  const int b  = blockIdx.y;
  const int tid = threadIdx.x;

  // Stage 64 Q rows (64x256 f32 = 64KB) into LDS via b128 loads/stores.
  for (int v = tid; v < (16 * KTILES) * (HID / 4); v += 224) {
    const int r  = v >> 6;            // HID/4 == 64 float4 chunks per row
    const int c4 = v & 63;
    const float4 q = *(const float4*)&query[(((size_t)k0 + r) * BSZ + b) * HID + c4 * 4];
    *(float4*)&ldsQ[r * QPAD + c4 * 4] = q;
  }
  __syncthreads();

  const int lane = tid & 31;
  const int wave = tid >> 5;
  const int hi   = lane >> 4;        // 0: lanes 0-15, 1: lanes 16-31
  const int ncol = lane & 15;        // N index within tile
  const int kb   = hi * 2;           // A/B K sub-offset per ISA layout
  const int n0   = wave * 16;

  // B fragment pointer: Mt2[b][c4][n0+ncol][kb..kb+1] -> contiguous b64 per step
  const float* mtp  = &Mt[(((size_t)b * (HID / 4)) * NPAD + n0 + ncol) * 4 + kb];
  const float* a0   = &ldsQ[(lane & 15) * QPAD + kb];

  v8f acc[KTILES];
#pragma unroll
  for (int kt = 0; kt < KTILES; ++kt) acc[kt] = (v8f){0.f,0.f,0.f,0.f,0.f,0.f,0.f,0.f};

#pragma unroll 4
  for (int c4 = 0; c4 < HID / 4; ++c4) {
    const v2f bb = *(const v2f*)(mtp + (size_t)c4 * NPAD * 4);   // global_load_b64
#pragma unroll
    for (int kt = 0; kt < KTILES; ++kt) {
      const v2f a = *(const v2f*)(a0 + kt * 16 * QPAD + c4 * 4); // ds_load_b64
      acc[kt] = __builtin_amdgcn_wmma_f32_16x16x4_f32(
          /*neg_a=*/false, a, /*neg_b=*/false, bb,
          /*c_mod=*/(short)0, acc[kt], /*reuse_a=*/false, /*reuse_b=*/false);
    }
  }

  const int n = n0 + ncol;
  if (n < NQ) {
    const float dv = dws[b * NQ + n];
#pragma unroll
    for (int kt = 0; kt < KTILES; ++kt) {
      const size_t idx = ((size_t)b * NQ + n) * HW + (k0 + kt * 16) + 8 * hi;
      float4 lo = make_float4(acc[kt][0] + dv, acc[kt][1] + dv, acc[kt][2] + dv, acc[kt][3] + dv);
      float4 hv = make_float4(acc[kt][4] + dv, acc[kt][5] + dv, acc[kt][6] + dv, acc[kt][7] + dv);
      *(float4*)(&sim[idx])     = lo;
      *(float4*)(&sim[idx + 4]) = hv;
    }
  }
}

// ---------------------------------------------------------------------------
// K4: per (b,n): softmax moments -> proposal_for_loss; argmax 3x3 window -> proposals
// grid (NQ, BS), block 256
// ---------------------------------------------------------------------------
__global__ void __launch_bounds__(256)
proposal_kernel(const float* __restrict__ sim,  // [BS,NQ,HW]
                float* __restrict__ pfl,        // [BS,NQ,2]
                float* __restrict__ props)      // [BS,NQ,2]
{
  __shared__ float row[HW];
  __shared__ float rv[256];
  __shared__ int   ri[256];
  __shared__ float s0[256], s1[256], s2[256];
  const int n = blockIdx.x, b = blockIdx.y, t = threadIdx.x;
  const float* r = &sim[((size_t)b * NQ + n) * HW];

  // pass 1: max + first-index argmax (matches jnp.argmax tie-break)
  float best = -INFINITY;
  int   bidx = HW;
  for (int k = t; k < HW; k += 256) {
    const float v = r[k];
    row[k] = v;
    if (v > best || (v == best && k < bidx)) { best = v; bidx = k; }
  }
  rv[t] = best; ri[t] = bidx;
  __syncthreads();
  for (int s = 128; s > 0; s >>= 1) {
    if (t < s) {
      if (rv[t + s] > rv[t] || (rv[t + s] == rv[t] && ri[t + s] < ri[t])) {
        rv[t] = rv[t + s]; ri[t] = ri[t + s];
      }
    }
    __syncthreads();
  }
  const float m  = rv[0];
  const int amax = ri[0];
  __syncthreads();

  // pass 2: softmax moments
  float S = 0.f, Sx = 0.f, Sy = 0.f;
  for (int k = t; k < HW; k += 256) {
    const float e = expf(row[k] - m);
    const int y = k / WW, x = k - y * WW;
    S += e;
    Sx = fmaf(e, (float)x + 0.5f, Sx);
    Sy = fmaf(e, (float)y + 0.5f, Sy);
  }
  s0[t] = S; s1[t] = Sx; s2[t] = Sy;
  __syncthreads();
  for (int s = 128; s > 0; s >>= 1) {
    if (t < s) { s0[t] += s0[t + s]; s1[t] += s1[t + s]; s2[t] += s2[t + s]; }
    __syncthreads();
  }

  if (t == 0) {
    const float St = s0[0], Sxt = s1[0], Syt = s2[0];
    const size_t o = ((size_t)b * NQ + n) * 2;
    pfl[o]     = Sxt / St / (float)WW;
    pfl[o + 1] = Syt / St / (float)HH;

    // 3x3 clipped window around argmax (h==w==80, so (w,h) reshape == (h,w))
    const int y0 = amax / WW, x0 = amax - y0 * WW;
    float E = 0.f, Ex = 0.f, Ey = 0.f;
    for (int dy = -1; dy <= 1; ++dy)
      for (int dx = -1; dx <= 1; ++dx) {
        const int y = y0 + dy, x = x0 + dx;
        if (y >= 0 && y < HH && x >= 0 && x < WW) {
          const float e = expf(row[y * WW + x] - m);
          E += e;
          Ex = fmaf(e, (float)x + 0.5f, Ex);
          Ey = fmaf(e, (float)y + 0.5f, Ey);
        }
      }
    // local_sm/(sum+1e-10) == e_i / (E + 1e-10*S_total)
    const float den = E + 1e-10f * St;
    props[o]     = Ex / den / (float)WW;
    props[o + 1] = Ey / den / (float)HH;
  }
}

// ---------------------------------------------------------------------------
extern "C" void kernel_launch(void* const* d_in, const int* in_sizes, int n_in,
                              void* d_out, int out_size, void* d_ws, size_t ws_size,
                              hipStream_t stream) {
  const float* query   = (const float*)d_in[0];   // [6400,16,256]
  const float* support = (const float*)d_in[1];   // [100,16,256]
  // d_in[2]=h, d_in[3]=w (known 80x80 from reference constants)
  const float* Wq  = (const float*)d_in[4];
  const float* bq  = (const float*)d_in[5];
  const float* Ws  = (const float*)d_in[6];
  const float* bsp = (const float*)d_in[7];
  const float* W1  = (const float*)d_in[8];
  const float* b1  = (const float*)d_in[9];
  const float* W2  = (const float*)d_in[10];
  const float* b2  = (const float*)d_in[11];

  float* out   = (float*)d_out;
  float* pfl   = out;                                 // [16,100,2]
  float* sim   = out + (size_t)BSZ * NQ * 2;          // [16,100,6400] (= similarity)
  float* props = sim + (size_t)BSZ * NQ * HW;         // [16,100,2]

  float* fsfeat = (float*)d_ws;                       // 409600 f32
  float* dws    = fsfeat + (size_t)BSZ * NQ * PROJ;   // 1600 f32
  float* Mt     = dws + (size_t)BSZ * NQ;             // 458752 f32 (~3.5MB total ws)

  fs_feat_kernel<<<dim3(NQ, BSZ), 256, 0, stream>>>(support, Ws, bsp, W1, b1, W2, b2,
                                                    bq, fsfeat, dws);
  mt_kernel<<<dim3(HID, BSZ), 128, 0, stream>>>(Wq, fsfeat, Mt);
  sim_wmma_kernel<<<dim3(HW / (16 * KTILES), BSZ), 224, 0, stream>>>(query, Mt, dws, sim);
  proposal_kernel<<<dim3(NQ, BSZ), 256, 0, stream>>>(sim, pfl, props);
}